// ScaledDotProductAttention_11081015623956
// MI455X (gfx1250) — compile-verified
//
#include <hip/hip_runtime.h>

// ---------------------------------------------------------------------------
// Scaled dot-product attention (fp32), CDNA5 / gfx1250, wave32.
//   context = softmax(mask(Q K^T * scale)) V      [B,H,L,D]
//   attn    = softmax probabilities               [B,H,L,L]
// d_out = context (B*H*L*D floats) followed by attn (B*H*L*L floats).
//
// Bandwidth-bound (~640 MB HBM traffic, 512 MB of it the attn store), so we
// use the exact-precision V_WMMA_F32_16X16X4_F32 path: fp32 fidelity at zero
// cost since compute is ~52 GFLOP total.
// ---------------------------------------------------------------------------

typedef __attribute__((ext_vector_type(2))) float v2f;
typedef __attribute__((ext_vector_type(8))) float v8f;

#define BSZ  2
#define HN   16
#define LSEQ 2048
#define DH   64
#define NW   8            // waves per block (256 threads, wave32)
#define QT   16           // query rows per block
#define KT   16           // key cols per tile
#define NKT  (LSEQ / KT)  // 128 k-tiles

__global__ __launch_bounds__(256)
void sdpa_fp32_wmma_kernel(const float* __restrict__ Q,
                           const float* __restrict__ K,
                           const float* __restrict__ V,
                           const float* __restrict__ scale_p,
                           const int*   __restrict__ mask,
                           float* __restrict__ ctx_out,
                           float* __restrict__ attn_out)
{
    const int qt   = blockIdx.x;            // 0..127
    const int h    = blockIdx.y;            // 0..15
    const int b    = blockIdx.z;            // 0..1
    const int lane = threadIdx.x & 31;
    const int wave = threadIdx.x >> 5;
    const int half = lane >> 4;             // C-matrix half: rows r+8*half
    const int idx  = lane & 15;             // column / A-row index

    const float scale = *scale_p;

    const size_t bh = (size_t)b * HN + h;
    const float* Qb = Q + (bh * LSEQ + (size_t)qt * QT) * DH;
    const float* Kb = K + bh * LSEQ * DH;
    const float* Vb = V + bh * LSEQ * DH;
    const int*   Mb = mask + ((size_t)b * LSEQ + (size_t)qt * QT) * LSEQ;
    float* Ob = ctx_out  + (bh * LSEQ + (size_t)qt * QT) * DH;
    float* Ab = attn_out + (bh * LSEQ + (size_t)qt * QT) * LSEQ;

    __shared__ float mstat[NW][16];
    __shared__ float lstat[NW][16];
    __shared__ __align__(16) float pbuf[NW][16][18];   // stride 18: 8B-aligned pairs, bank-spread
    __shared__ float ctxbuf[16][64];

    // ---- Preload Q tile as WMMA A fragments (16x4 f32 per chunk) ----------
    // A layout: lanes 0-15 / 16-31 both cover M=0..15; VGPR0 = {K, K+2}, VGPR1 = {K+1, K+3}.
    // Per lane (half,idx): a = Q[idx][4*kc + 2*half .. +1]
    v2f qa[16];
#pragma unroll
    for (int kc = 0; kc < 16; ++kc)
        qa[kc] = *(const v2f*)(Qb + (size_t)idx * DH + 4 * kc + 2 * half);

    // S = Q * K^T for one 16x16 tile: 16 chained f32 WMMAs over D=64.
    // B operand of Q*K^T has the SAME per-lane addressing as A on K's rows.
    auto computeS = [&](int kt) -> v8f {
        v8f c = {};
        const float* Kt = Kb + (size_t)(kt * KT + idx) * DH + 2 * half;
#pragma unroll
        for (int kc = 0; kc < 16; ++kc) {
            v2f kb = *(const v2f*)(Kt + 4 * kc);
            c = __builtin_amdgcn_wmma_f32_16x16x4_f32(
                    false, qa[kc], false, kb, (short)0, c, false, false);
        }
        return c;
    };

    // ---- Pass 1: per-wave online softmax stats (row max, row sumexp) ------
    float m_loc[8], l_loc[8];
#pragma unroll
    for (int r = 0; r < 8; ++r) { m_loc[r] = -__builtin_inff(); l_loc[r] = 0.f; }

    for (int kt = wave; kt < NKT; kt += NW) {
        if (kt + NW < NKT)  // prefetch next K tile (global_prefetch_b8)
            __builtin_prefetch(Kb + (size_t)((kt + NW) * KT + idx) * DH + 2 * half, 0, 0);

        v8f c = computeS(kt);
        float s[8];
#pragma unroll
        for (int r = 0; r < 8; ++r) {
            const int mrow = r + 8 * half;
            const int mi = Mb[(size_t)mrow * LSEQ + kt * KT + idx];
            const float sv = c[r] * scale;
            s[r] = mi ? -1.0e6f : sv;   // == s*(1-m) + m*(-1e6) for m in {0,1}
        }
        // row max / sumexp across the 16 lanes of each half (wave32 shuffles)
#pragma unroll
        for (int r = 0; r < 8; ++r) {
            float v = s[r];
            v = fmaxf(v, __shfl_xor(v, 1, 32));
            v = fmaxf(v, __shfl_xor(v, 2, 32));
            v = fmaxf(v, __shfl_xor(v, 4, 32));
            v = fmaxf(v, __shfl_xor(v, 8, 32));
            float e = __expf(s[r] - v);
            e += __shfl_xor(e, 1, 32);
            e += __shfl_xor(e, 2, 32);
            e += __shfl_xor(e, 4, 32);
            e += __shfl_xor(e, 8, 32);
            const float nm = fmaxf(m_loc[r], v);
            l_loc[r] = l_loc[r] * __expf(m_loc[r] - nm) + e * __expf(v - nm);
            m_loc[r] = nm;
        }
    }

    // ---- Cross-wave (m, l) merge through LDS ------------------------------
    if (idx == 0) {
#pragma unroll
        for (int r = 0; r < 8; ++r) {
            mstat[wave][r + 8 * half] = m_loc[r];
            lstat[wave][r + 8 * half] = l_loc[r];
        }
    }
    __syncthreads();

    float m_g[8], inv_l[8];
#pragma unroll
    for (int r = 0; r < 8; ++r) {
        const int row = r + 8 * half;
        float mg = -__builtin_inff();
        for (int w = 0; w < NW; ++w) mg = fmaxf(mg, mstat[w][row]);
        float lg = 0.f;
        for (int w = 0; w < NW; ++w) lg += lstat[w][row] * __expf(mstat[w][row] - mg);
        m_g[r]   = mg;
        inv_l[r] = 1.0f / lg;
    }

    // zero the cross-wave context accumulator
    for (int t = threadIdx.x; t < 16 * 64; t += 256)
        ((float*)ctxbuf)[t] = 0.f;
    __syncthreads();

    // ---- Pass 2: normalized attn store + P*V accumulation -----------------
    v8f ctx[4] = {};
    for (int kt = wave; kt < NKT; kt += NW) {
        if (kt + NW < NKT)
            __builtin_prefetch(Kb + (size_t)((kt + NW) * KT + idx) * DH + 2 * half, 0, 0);

        v8f c = computeS(kt);
#pragma unroll
        for (int r = 0; r < 8; ++r) {
            const int mrow = r + 8 * half;
            const int mi = Mb[(size_t)mrow * LSEQ + kt * KT + idx];
            float sv = c[r] * scale;
            sv = mi ? -1.0e6f : sv;
            const float p = __expf(sv - m_g[r]) * inv_l[r];
            Ab[(size_t)mrow * LSEQ + kt * KT + idx] = p;   // attn output (written once)
            pbuf[wave][mrow][idx] = p;                     // relayout for A operand
        }
        // P (16x16) @ V_tile (16x64): 4 k-chunks x 4 n-tiles of f32 WMMA.
        // Same-wave LDS write->read: DS ops are in-order per wave; compiler waits dscnt.
#pragma unroll
        for (int kc = 0; kc < 4; ++kc) {
            const v2f pa = *(const v2f*)&pbuf[wave][idx][4 * kc + 2 * half];
            const int krow = kt * KT + 4 * kc + 2 * half;
#pragma unroll
            for (int vt = 0; vt < 4; ++vt) {
                v2f vb;
                vb.x = Vb[(size_t)krow * DH + vt * 16 + idx];
                vb.y = Vb[(size_t)(krow + 1) * DH + vt * 16 + idx];
                ctx[vt] = __builtin_amdgcn_wmma_f32_16x16x4_f32(
                              false, pa, false, vb, (short)0, ctx[vt], false, false);
            }
        }
    }

    // ---- Cross-wave context reduction (ds_add_f32) and store --------------
#pragma unroll
    for (int vt = 0; vt < 4; ++vt)
#pragma unroll
        for (int r = 0; r < 8; ++r)
            atomicAdd(&ctxbuf[r + 8 * half][vt * 16 + idx], ctx[vt][r]);
    __syncthreads();

    for (int t = threadIdx.x; t < 16 * 64; t += 256) {
        const int row = t >> 6, col = t & 63;
        Ob[(size_t)row * DH + col] = ctxbuf[row][col];
    }
}

extern "C" void kernel_launch(void* const* d_in, const int* in_sizes, int n_in,
                              void* d_out, int out_size, void* d_ws, size_t ws_size,
                              hipStream_t stream) {
    (void)in_sizes; (void)n_in; (void)out_size; (void)d_ws; (void)ws_size;
    const float* Q     = (const float*)d_in[0];
    const float* K     = (const float*)d_in[1];
    const float* V     = (const float*)d_in[2];
    const float* scale = (const float*)d_in[3];   // 1-element device array
    const int*   mask  = (const int*)d_in[4];

    float* ctx_out  = (float*)d_out;                                  // [B,H,L,D]
    float* attn_out = ctx_out + (size_t)BSZ * HN * LSEQ * DH;         // [B,H,L,L]

    dim3 grid(LSEQ / QT, HN, BSZ);   // (128, 16, 2)
    dim3 block(256);                 // 8 wave32 waves
    sdpa_fp32_wmma_kernel<<<grid, block, 0, stream>>>(
        Q, K, V, scale, mask, ctx_out, attn_out);
}